// JointBlockEmbedder_8907762172482
// MI455X (gfx1250) — compile-verified
//
#include <hip/hip_runtime.h>
#include <stdint.h>

// ---------------------------------------------------------------------------
// JointBlockEmbedder for MI455X (gfx1250, wave32, WMMA + async global->LDS).
//
// Input order (setup_inputs dict order):
//  0 block_embs (B,N,6)      1 spans (B,N,3)
//  2 content_w1(16,6)  3 content_b1(16)  4 content_w2(64,16)  5 content_b2(64)
//  6 skip_w(64,6)      7 skip_b(64)      8 chr_emb(23,8)
//  9 film_w1(16,40) 10 film_b1(16) 11 film_w2(128,16) 12 film_b2(128)
// 13 jit_w1(16,40) 14 jit_b1(16) 15 jit_w2(2,16) 16 jit_b2(2)
// 17 refine_w1(64,129,3,3) 18 refine_b1(64) 19 refine_w2(64,64,3,3) 20 refine_b2(64)
// 21 pre_w1(128,64,1,1) 22 pre_b1(128) 23 pre_w2(64,128,1,1) 24 pre_b2(64)
// 25 lat_w(128,64,1,1) 26 lat_b(128) 27 base_coords(N,2)
// ---------------------------------------------------------------------------

typedef __attribute__((ext_vector_type(2))) float v2f;
typedef __attribute__((ext_vector_type(8))) float v8f;

namespace cfg {
constexpr int B = 8, N = 65536, D = 6, E = 64, POS = 16, LAT = 128;
constexpr int H = 128, W = 128, HW = H * W;
constexpr int NFREQ = 8, CHR = 8, FIN = CHR + 4 * NFREQ; // 40
constexpr float JS = 1.5f / 128.0f;
constexpr float TWO_PI = 6.2831853071795864769f;
}

__device__ __forceinline__ float silu_f(float x) { return x / (1.f + __expf(-x)); }

// Async copy of one dword global -> LDS (CDNA5 GLOBAL_LOAD_ASYNC_TO_LDS_B32,
// tracked by ASYNCcnt). ldsOff = byte offset within the wave's LDS allocation
// (low 32 bits of a generic pointer to __shared__ memory).
__device__ __forceinline__ void async_g2l_b32(unsigned ldsOff, const float* gp) {
  asm volatile("global_load_async_to_lds_b32 %0, %1, off"
               :: "v"(ldsOff), "v"((unsigned long long)(uintptr_t)gp)
               : "memory");
}
__device__ __forceinline__ void wait_asynccnt0() {
  asm volatile("s_wait_asynccnt 0x0" ::: "memory");
}

// --------------------------------------------------------------------------
// Zero scratch grids (F + M are contiguous in ws).
// --------------------------------------------------------------------------
__global__ void zero_kernel(float* __restrict__ p, int n) {
  int i = blockIdx.x * 256 + threadIdx.x;
  if (i < n) p[i] = 0.f;
}

// --------------------------------------------------------------------------
// Repack OIHW conv weights -> [k = c*KH*KW + r*KW + s][o]  (GEMM-B layout).
// src index = o*(C*KK) + (c*KK + rs)  => dst[(c*KK+rs)*O + o] = src[idx]
// --------------------------------------------------------------------------
__global__ void repack_kernel(const float* __restrict__ src, float* __restrict__ dst,
                              int O, int CKK) {
  int idx = blockIdx.x * 256 + threadIdx.x;
  int total = O * CKK;
  if (idx >= total) return;
  int o = idx / CKK;
  int rem = idx - o * CKK;
  dst[rem * O + o] = src[idx];
}

// --------------------------------------------------------------------------
// Fused per-point stage: content MLP + FiLM + jitter MLP + bilinear splat.
// One thread per (b,n) point; all loops fully unrolled so private arrays
// live in VGPRs. Weights are wave-uniform -> scalar (constant-cache) loads.
// Splat: global_atomic_add_f32 into L2-resident F/M grids.
// --------------------------------------------------------------------------
__global__ __launch_bounds__(256) void point_kernel(
    const float* __restrict__ emb, const float* __restrict__ spans,
    const float* __restrict__ cw1, const float* __restrict__ cb1,
    const float* __restrict__ cw2, const float* __restrict__ cb2,
    const float* __restrict__ skw, const float* __restrict__ skb,
    const float* __restrict__ chrE,
    const float* __restrict__ fw1, const float* __restrict__ fb1,
    const float* __restrict__ fw2, const float* __restrict__ fb2,
    const float* __restrict__ jw1, const float* __restrict__ jb1,
    const float* __restrict__ jw2, const float* __restrict__ jb2,
    const float* __restrict__ base,
    float* __restrict__ Fg, float* __restrict__ Mg) {
  using namespace cfg;
  int gid = blockIdx.x * 256 + threadIdx.x;
  if (gid >= B * N) return;
  int b = gid / N;
  int n = gid - b * N;

  float eb[D];
#pragma unroll
  for (int d = 0; d < D; ++d) eb[d] = emb[(long)gid * D + d];

  const float* sp = &spans[(long)gid * 3];
  int ci = (int)sp[0];
  float st = sp[1], ln = sp[2];

  // film_in = [chr_emb(8), sin(2pi s f), cos(2pi s f), sin(2pi l f), cos(2pi l f)]
  float fin[FIN];
#pragma unroll
  for (int i = 0; i < CHR; ++i) fin[i] = chrE[ci * CHR + i];
#pragma unroll
  for (int f = 0; f < NFREQ; ++f) {
    float fr = (float)(1 << f);
    float as = TWO_PI * st * fr, al = TWO_PI * ln * fr;
    fin[CHR + f] = __sinf(as);
    fin[CHR + NFREQ + f] = __cosf(as);
    fin[CHR + 2 * NFREQ + f] = __sinf(al);
    fin[CHR + 3 * NFREQ + f] = __cosf(al);
  }

  // content hidden (silu)
  float h[POS];
#pragma unroll
  for (int p = 0; p < POS; ++p) {
    float a = cb1[p];
#pragma unroll
    for (int d = 0; d < D; ++d) a += cw1[p * D + d] * eb[d];
    h[p] = silu_f(a);
  }

  // FiLM hidden (leaky 0.2)
  float fh[POS];
#pragma unroll
  for (int p = 0; p < POS; ++p) {
    float a = fb1[p];
#pragma unroll
    for (int i = 0; i < FIN; ++i) a += fw1[p * FIN + i] * fin[i];
    fh[p] = a > 0.f ? a : 0.2f * a;
  }

  // jitter hidden (leaky 0.2) -> delta
  float jh[POS];
#pragma unroll
  for (int p = 0; p < POS; ++p) {
    float a = jb1[p];
#pragma unroll
    for (int i = 0; i < FIN; ++i) a += jw1[p * FIN + i] * fin[i];
    jh[p] = a > 0.f ? a : 0.2f * a;
  }
  float d0 = jb2[0], d1 = jb2[1];
#pragma unroll
  for (int p = 0; p < POS; ++p) {
    d0 += jw2[p] * jh[p];
    d1 += jw2[POS + p] * jh[p];
  }
  d0 = tanhf(d0) * JS;
  d1 = tanhf(d1) * JS;

  float u = fminf(fmaxf(base[n * 2 + 0] + d0, 0.f), 1.f);
  float v = fminf(fmaxf(base[n * 2 + 1] + d1, 0.f), 1.f);
  float du = u * (float)(W - 1), dv = v * (float)(H - 1);
  int j0 = (int)floorf(du), i0 = (int)floorf(dv);
  int j1 = min(j0 + 1, W - 1), i1 = min(i0 + 1, H - 1);
  float j0f = (float)j0, i0f = (float)i0, j1f = (float)j1, i1f = (float)i1;
  float w00 = (i1f - dv) * (j1f - du);
  float w01 = (i1f - dv) * (du - j0f);
  float w10 = (dv - i0f) * (j1f - du);
  float w11 = (dv - i0f) * (du - j0f);
  int p00 = i0 * W + j0, p01 = i0 * W + j1, p10 = i1 * W + j0, p11 = i1 * W + j1;

  float* Mb = Mg + (long)b * HW;
  atomicAdd(Mb + p00, w00);
  atomicAdd(Mb + p01, w01);
  atomicAdd(Mb + p10, w10);
  atomicAdd(Mb + p11, w11);

  float* Fb = Fg + (long)b * E * HW;
#pragma unroll
  for (int e = 0; e < E; ++e) {
    // content channel e (fused: w2 @ h + skip @ emb)
    float ce = cb2[e] + skb[e];
#pragma unroll
    for (int p = 0; p < POS; ++p) ce += cw2[e * POS + p] * h[p];
#pragma unroll
    for (int d = 0; d < D; ++d) ce += skw[e * D + d] * eb[d];
    // FiLM gamma/beta for channel e
    float ga = fb2[e], be = fb2[E + e];
#pragma unroll
    for (int p = 0; p < POS; ++p) {
      ga += fw2[e * POS + p] * fh[p];
      be += fw2[(E + e) * POS + p] * fh[p];
    }
    float xe = (1.f + 0.5f * tanhf(ga)) * ce + be;
    float* Fe = Fb + (long)e * HW;
    atomicAdd(Fe + p00, xe * w00);
    atomicAdd(Fe + p01, xe * w01);
    atomicAdd(Fe + p10, xe * w10);
    atomicAdd(Fe + p11, xe * w11);
  }
}

// --------------------------------------------------------------------------
// Implicit-GEMM convolution on V_WMMA_F32_16X16X4_F32.
// One wave per 16-pixel strip (b, y, x0..x0+15). M=16 pixels, N=COUT (tiled
// by 16), K=CIN*KS*KS stepped by 4. Input patch staged in LDS; for plain
// grid inputs the staging uses CDNA5 async global->LDS DMA (ASYNCcnt);
// conv1 synthesizes the concat(mu, F, M) 129-channel input on the fly.
// A frag (ISA 7.12.2, 32-bit A 16x4): lane<16 -> K={kb,kb+1}, lane>=16 ->
// K={kb+2,kb+3}, lane&15 = M. B frag mirrors the K split with lane&15 = N.
// D: lanes = N, VGPR v = M%8, lane-half = M/8 -> 8 consecutive-x stores.
// --------------------------------------------------------------------------
template <int CIN, int COUT, int KS, bool SILU, bool CONCAT>
__global__ __launch_bounds__(32) void conv_wmma_kernel(
    const float* __restrict__ in,   // (B,CIN,H,W) when !CONCAT
    const float* __restrict__ Fg,   // (B,64,H,W)  when CONCAT
    const float* __restrict__ Mg,   // (B,H,W)     when CONCAT
    const float* __restrict__ wr,   // [K][COUT] repacked weights
    const float* __restrict__ bias, // [COUT]
    float* __restrict__ out) {      // (B,COUT,H,W)
  using namespace cfg;
  constexpr int KK = KS * KS;
  constexpr int K = CIN * KK;
  constexpr int PR = (KS == 3) ? 3 : 1;
  constexpr int PW = (KS == 3) ? 18 : 16;
  constexpr int NT = COUT / 16;

  __shared__ float patch[CIN * PR * PW];

  const int lane = threadIdx.x;
  const int strip = blockIdx.x;               // H * (W/16) strips
  const int b = blockIdx.y;
  const int y = strip / (W / 16);
  const int x0 = (strip - y * (W / 16)) * 16;

  __builtin_prefetch(wr, 0, 1);  // global_prefetch_b8: weights are L2-hot

  // ---- stage zero-padded input patch into LDS -------------------------
  constexpr int TOTAL = CIN * PR * PW;
  for (int i = lane; i < TOTAL; i += 32) {
    int c = i / (PR * PW);
    int rem = i - c * (PR * PW);
    int r = rem / PW;
    int xx = rem - r * PW;
    int gy = y + r - (KS == 3 ? 1 : 0);
    int gx = x0 + xx - (KS == 3 ? 1 : 0);
    bool inb = (gy >= 0 && gy < H && gx >= 0 && gx < W);
    if constexpr (CONCAT) {
      float val = 0.f;
      if (inb) {
        int p = gy * W + gx;
        float m = Mg[b * HW + p];
        if (c < 64)
          val = Fg[((b * 64 + c) * HW) + p] / (m + 1e-8f);
        else if (c < 128)
          val = Fg[((b * 64 + (c - 64)) * HW) + p];
        else
          val = m;
      }
      patch[i] = val;
    } else {
      if (inb) {
        // CDNA5 async DMA: global -> LDS without touching VGPR data paths.
        unsigned ldsOff = (unsigned)(unsigned long long)(uintptr_t)&patch[i];
        async_g2l_b32(ldsOff, &in[((long)(b * CIN + c) * H + gy) * W + gx]);
      } else {
        patch[i] = 0.f;  // zero-pad halo via plain ds_store
      }
    }
  }
  if constexpr (!CONCAT) wait_asynccnt0();  // drain ASYNCcnt before reading LDS
  __syncthreads();

  // ---- WMMA main loop --------------------------------------------------
  v8f acc[NT] = {};
  const int m = lane & 15;
  const int khalf = (lane >> 4) * 2;
  const int nlo = lane & 15;

  auto ldA = [&](int k) -> float {
    if constexpr (KS == 3) {
      int c = k / 9;
      int rm = k - c * 9;
      int r = rm / 3;
      int s = rm - r * 3;
      return patch[(c * 3 + r) * 18 + m + s];
    } else {
      return patch[k * 16 + m];
    }
  };

#pragma unroll 2
  for (int kb = 0; kb < K; kb += 4) {
    int k0 = kb + khalf;
    int k1 = k0 + 1;
    bool v0 = (k0 < K), v1 = (k1 < K);
    int k0c = v0 ? k0 : 0, k1c = v1 ? k1 : 0;
    v2f a;
    a.x = v0 ? ldA(k0c) : 0.f;
    a.y = v1 ? ldA(k1c) : 0.f;
#pragma unroll
    for (int t = 0; t < NT; ++t) {
      v2f bf;
      bf.x = v0 ? wr[k0c * COUT + t * 16 + nlo] : 0.f;
      bf.y = v1 ? wr[k1c * COUT + t * 16 + nlo] : 0.f;
      acc[t] = __builtin_amdgcn_wmma_f32_16x16x4_f32(
          /*neg_a=*/false, a, /*neg_b=*/false, bf,
          /*c_mod=*/(short)0, acc[t], /*reuse_a=*/false, /*reuse_b=*/false);
    }
  }

  // ---- epilogue: bias (+silu), store 8 consecutive x per lane ---------
  const int xbase = x0 + ((lane >> 4) ? 8 : 0);
#pragma unroll
  for (int t = 0; t < NT; ++t) {
    int ch = t * 16 + nlo;
    float bb = bias[ch];
    float* op = &out[((long)(b * COUT + ch) * H + y) * W + xbase];
#pragma unroll
    for (int v = 0; v < 8; ++v) {
      float r = acc[t][v] + bb;
      if constexpr (SILU) r = silu_f(r);
      op[v] = r;
    }
  }
}

// --------------------------------------------------------------------------
extern "C" void kernel_launch(void* const* d_in, const int* in_sizes, int n_in,
                              void* d_out, int out_size, void* d_ws, size_t ws_size,
                              hipStream_t stream) {
  using namespace cfg;
  const float* block_embs = (const float*)d_in[0];
  const float* spans = (const float*)d_in[1];
  const float* cw1 = (const float*)d_in[2];
  const float* cb1 = (const float*)d_in[3];
  const float* cw2 = (const float*)d_in[4];
  const float* cb2 = (const float*)d_in[5];
  const float* skw = (const float*)d_in[6];
  const float* skb = (const float*)d_in[7];
  const float* chrE = (const float*)d_in[8];
  const float* fw1 = (const float*)d_in[9];
  const float* fb1 = (const float*)d_in[10];
  const float* fw2 = (const float*)d_in[11];
  const float* fb2 = (const float*)d_in[12];
  const float* jw1 = (const float*)d_in[13];
  const float* jb1 = (const float*)d_in[14];
  const float* jw2 = (const float*)d_in[15];
  const float* jb2 = (const float*)d_in[16];
  const float* refine_w1 = (const float*)d_in[17];
  const float* refine_b1 = (const float*)d_in[18];
  const float* refine_w2 = (const float*)d_in[19];
  const float* refine_b2 = (const float*)d_in[20];
  const float* pre_w1 = (const float*)d_in[21];
  const float* pre_b1 = (const float*)d_in[22];
  const float* pre_w2 = (const float*)d_in[23];
  const float* pre_b2 = (const float*)d_in[24];
  const float* lat_w = (const float*)d_in[25];
  const float* lat_b = (const float*)d_in[26];
  const float* base = (const float*)d_in[27];

  // workspace layout (floats)
  float* ws = (float*)d_ws;
  size_t off = 0;
  float* Fgrid = ws + off; off += (size_t)B * E * HW;   // 8.39M
  float* Mgrid = ws + off; off += (size_t)B * HW;       // 131K (contiguous after F)
  float* gA = ws + off;    off += (size_t)B * E * HW;   // refine1 out / pre2 out
  float* gB = ws + off;    off += (size_t)B * E * HW;   // refine2 out
  float* pH = ws + off;    off += (size_t)B * 128 * HW; // pre1 hidden (128ch)
  float* wr1 = ws + off;   off += (size_t)129 * 9 * 64;
  float* wr2 = ws + off;   off += (size_t)64 * 9 * 64;
  float* wp1 = ws + off;   off += (size_t)64 * 128;
  float* wp2 = ws + off;   off += (size_t)128 * 64;
  float* wl = ws + off;    off += (size_t)64 * 128;

  // 1) zero accumulation grids (F and M are contiguous)
  {
    int n = B * E * HW + B * HW;
    zero_kernel<<<(n + 255) / 256, 256, 0, stream>>>(Fgrid, n);
  }
  // 2) repack conv weights OIHW -> [k][o]
  repack_kernel<<<(64 * 129 * 9 + 255) / 256, 256, 0, stream>>>(refine_w1, wr1, 64, 129 * 9);
  repack_kernel<<<(64 * 64 * 9 + 255) / 256, 256, 0, stream>>>(refine_w2, wr2, 64, 64 * 9);
  repack_kernel<<<(128 * 64 + 255) / 256, 256, 0, stream>>>(pre_w1, wp1, 128, 64);
  repack_kernel<<<(64 * 128 + 255) / 256, 256, 0, stream>>>(pre_w2, wp2, 64, 128);
  repack_kernel<<<(128 * 64 + 255) / 256, 256, 0, stream>>>(lat_w, wl, 128, 64);

  // 3) fused per-point MLPs + bilinear splat
  point_kernel<<<(B * N + 255) / 256, 256, 0, stream>>>(
      block_embs, spans, cw1, cb1, cw2, cb2, skw, skb, chrE, fw1, fb1, fw2, fb2,
      jw1, jb1, jw2, jb2, base, Fgrid, Mgrid);

  // 4) conv stack on WMMA f32 (one wave per 16-pixel strip)
  dim3 cgrid(H * (W / 16), B);
  conv_wmma_kernel<129, 64, 3, true, true><<<cgrid, 32, 0, stream>>>(
      nullptr, Fgrid, Mgrid, wr1, refine_b1, gA);                 // refine1 + silu
  conv_wmma_kernel<64, 64, 3, false, false><<<cgrid, 32, 0, stream>>>(
      gA, nullptr, nullptr, wr2, refine_b2, gB);                  // refine2
  conv_wmma_kernel<64, 128, 1, true, false><<<cgrid, 32, 0, stream>>>(
      gB, nullptr, nullptr, wp1, pre_b1, pH);                     // pre1 + silu
  conv_wmma_kernel<128, 64, 1, false, false><<<cgrid, 32, 0, stream>>>(
      pH, nullptr, nullptr, wp2, pre_b2, gA);                     // pre2
  conv_wmma_kernel<64, 128, 1, false, false><<<cgrid, 32, 0, stream>>>(
      gA, nullptr, nullptr, wl, lat_b, (float*)d_out);            // lat -> output
}